// Model_74663711473751
// MI455X (gfx1250) — compile-verified
//
#include <hip/hip_runtime.h>

typedef __attribute__((ext_vector_type(2))) float v2f;
typedef __attribute__((ext_vector_type(8))) float v8f;

// ---------------------------------------------------------------------------
// Kernel 1: build Psi (96x8, row-major) in workspace.
// Phi[t][k] = w[k-t] (if t<=k<8)  +  sum_{j=max(0,8-t)}^{7} w[j]*Phi[t+j-8][k]
// Psi[t][k] = sum_{s<=t} Phi[s][k]
// Columns k are independent -> one lane per k (8 active lanes, trivial work).
// ---------------------------------------------------------------------------
__global__ void compute_psi_kernel(const float* __restrict__ w_ar,
                                   float* __restrict__ psi /*96*8 floats*/) {
    const int k = threadIdx.x;
    if (k >= 8) return;
    float w[8];
#pragma unroll
    for (int j = 0; j < 8; ++j) w[j] = w_ar[j];

    float phi[96];
    float acc = 0.0f;
    for (int t = 0; t < 96; ++t) {
        float v = (t <= k) ? w[k - t] : 0.0f;   // direct term (k<8 always)
        int jstart = 8 - t; if (jstart < 0) jstart = 0;
#pragma unroll
        for (int j = 0; j < 8; ++j) {
            if (j >= jstart) v += w[j] * phi[t + j - 8];
        }
        phi[t] = v;
        acc += v;
        psi[t * 8 + k] = acc;                   // row-cumsum on the fly
    }
}

// ---------------------------------------------------------------------------
// Kernel 2: Out(96 x 4096) = Psi(96x8) x D(8x4096) + xlast, via
// V_WMMA_F32_16X16X4_F32. One wave per 16-column tile (256 waves total).
// Each wave: 6 M-tiles x 2 chained K=4 WMMAs = 12 WMMA ops.
//
// B-fragment layout for 4x16 f32 (2 VGPRs):
//   VGPR0: lanes 0-15 -> K=0, lanes 16-31 -> K=2
//   VGPR1: lanes 0-15 -> K=1, lanes 16-31 -> K=3
// => a lane only needs diffs {kb, kb+1, kb+4, kb+5} with kb = 2*(lane>=16).
// Fold kb into the *addresses* (no dynamic register indexing -> no cndmasks).
// ---------------------------------------------------------------------------
__global__ void arima_wmma_kernel(const float* __restrict__ x_enc,
                                  const float* __restrict__ psi,
                                  float* __restrict__ out) {
    constexpr int S = 4096, C = 64, T = 96;

    const int lane  = threadIdx.x & 31;
    const int wave  = blockIdx.x * (blockDim.x >> 5) + (threadIdx.x >> 5); // 0..255
    const int n0    = wave * 16;          // first flat (b,c) column of this tile
    const int half  = lane >> 4;          // 0: lanes 0-15, 1: lanes 16-31
    const int l15   = lane & 15;
    const int ncol  = n0 + l15;           // column this lane services
    const int b     = ncol >> 6;
    const int c     = ncol & 63;
    const int kb    = half * 2;           // K-subgroup handled by this lane

    // x_enc[b, S-9 .. S-1, c]; this lane needs samples kb..kb+2, kb+4..kb+6, 8.
    const float* xp = x_enc + (size_t)b * S * C + (size_t)(S - 9) * C + c
                            + (size_t)kb * C;
    const float x0 = xp[0 * C];
    const float x1 = xp[1 * C];
    const float x2 = xp[2 * C];
    const float x4 = xp[4 * C];
    const float x5 = xp[5 * C];
    const float x6 = xp[6 * C];
    const float xlast = xp[(8 - kb) * C];          // x_enc[b, S-1, c]

    v2f bfrag0; bfrag0.x = x1 - x0; bfrag0.y = x2 - x1;   // d[kb],   d[kb+1]
    v2f bfrag1; bfrag1.x = x5 - x4; bfrag1.y = x6 - x5;   // d[kb+4], d[kb+5]

    // 6 M-tiles cover all 96 output timesteps.
    const float* psiRowBase = psi + l15 * 8 + kb;   // + m0*8 (+4 for frag1)
#pragma unroll
    for (int mt = 0; mt < 6; ++mt) {
        const int m0 = mt * 16;
        v8f acc = {};
        {   // K = 0..3 chunk
            const v2f afrag = *(const v2f*)(psiRowBase + m0 * 8);
            acc = __builtin_amdgcn_wmma_f32_16x16x4_f32(
                      false, afrag, false, bfrag0, (short)0, acc, false, false);
        }
        {   // K = 4..7 chunk
            const v2f afrag = *(const v2f*)(psiRowBase + m0 * 8 + 4);
            acc = __builtin_amdgcn_wmma_f32_16x16x4_f32(
                      false, afrag, false, bfrag1, (short)0, acc, false, false);
        }
        // C/D layout: VGPR v -> row t = m0 + v + 8*half, col = l15 -> ncol
#pragma unroll
        for (int v = 0; v < 8; ++v) {
            const int t = m0 + v + half * 8;
            out[(size_t)b * T * C + (size_t)t * C + c] = acc[v] + xlast;
        }
    }
}

// ---------------------------------------------------------------------------
extern "C" void kernel_launch(void* const* d_in, const int* in_sizes, int n_in,
                              void* d_out, int out_size, void* d_ws, size_t ws_size,
                              hipStream_t stream) {
    (void)in_sizes; (void)n_in; (void)out_size; (void)ws_size;
    const float* x_enc = (const float*)d_in[0];   // (64, 4096, 64) f32
    const float* w_ar  = (const float*)d_in[4];   // (8,) f32   (w_ma unused: MA term == 0)
    float*       psi   = (float*)d_ws;            // 96*8 floats = 3 KB scratch
    float*       out   = (float*)d_out;           // (64, 96, 64) f32

    compute_psi_kernel<<<1, 32, 0, stream>>>(w_ar, psi);
    // 256 waves: 32 blocks x 256 threads (8 waves/block, wave32)
    arima_wmma_kernel<<<32, 256, 0, stream>>>(x_enc, psi, out);
}